// NNstack_88880053223465
// MI455X (gfx1250) — compile-verified
//
#include <hip/hip_runtime.h>

// ---------------------------------------------------------------------------
// MI455X plan:
//   Cost = Val copy: 134MB rd + 134MB wr = 268MB -> ~11.5us @ 23.3TB/s.
//   Kernel B moves tiles with the Tensor Data Mover (TENSOR_LOAD_TO_LDS /
//   TENSOR_STORE_FROM_LDS, TENSORcnt-tracked, double-buffered): 32x128 f32
//   tiles give 512B-contiguous bursts per row instead of 64B per-lane
//   segments, and the copy-out is a pure LDS->global DMA. WMMA
//   (V_WMMA_F32_16X16X4_F32, full f32 precision) consumes the same LDS tile
//   to accumulate rt partials, so the matvec adds zero DRAM traffic.
//   Scan kernel exploits the absorbing state u<=0 (then stg_i == max(0,s_i)
//   exactly), then a block-wide Hillis-Steele scan builds coef.
// ---------------------------------------------------------------------------

#define T_PREV 4095
#define NROWS  4096
#define MCOLS  8192
#define KSPLIT 8
#define ROWS_PER_SPLIT (NROWS / KSPLIT)   // 512
#define TILE_R 32
#define TILE_C 128
#define NCHUNK (ROWS_PER_SPLIT / TILE_R)  // 16

typedef __attribute__((ext_vector_type(2))) float v2f;
typedef __attribute__((ext_vector_type(8))) float v8f;
typedef __attribute__((ext_vector_type(4))) unsigned int v4u;
typedef __attribute__((ext_vector_type(4))) int v4i;
typedef __attribute__((ext_vector_type(8))) int v8i;

#ifndef __has_builtin
#define __has_builtin(x) 0
#endif
#if __has_builtin(__builtin_amdgcn_tensor_load_to_lds)
#define USE_TDM 1
#else
#define USE_TDM 0
#endif
#if __has_include(<hip/amd_detail/amd_gfx1250_TDM.h>)
#define TDM_6ARG 1   // therock-10.0 headers -> 6-arg builtin
#else
#define TDM_6ARG 0   // ROCm 7.2 clang-22 -> 5-arg builtin
#endif

// ---------------------------------------------------------------------------
// Kernel A: sequential pop-scan + reverse-cumsum -> stg (to out) and coef (ws)
// ---------------------------------------------------------------------------
__global__ void __launch_bounds__(1024) scan_coef_kernel(
    const float* __restrict__ prev_stg, const float* __restrict__ dt,
    const float* __restrict__ ut, float* __restrict__ out_stg,
    float* __restrict__ coef) {
  __shared__ float sstg[NROWS];
  __shared__ float sscan[1024];
  __shared__ int scross;
  const int tid = threadIdx.x;

  for (int i = tid; i < T_PREV; i += 1024) sstg[i] = prev_stg[i];
  if (tid == 0) sstg[T_PREV] = dt[0];
  __syncthreads();

  // Serial recurrence over decreasing i. u<=0 is absorbing: afterwards
  // stg_i = max(0, s_i) exactly -> finish in parallel.
  if (tid == 0) {
    float u = ut[0];
    int cross = 0;
    for (int i = T_PREV - 1; i >= 0; --i) {
      float g = fmaxf(0.0f, sstg[i] - fmaxf(0.0f, u));
      sstg[i] = g;
      u -= g;
      if (u <= 0.0f) { cross = i; break; }  // entries [0, i) still raw
    }
    scross = cross;
  }
  __syncthreads();
  const int cross = scross;
  for (int i = tid; i < cross; i += 1024) sstg[i] = fmaxf(0.0f, sstg[i]);
  __syncthreads();

  for (int i = tid; i < NROWS; i += 1024) out_stg[i] = sstg[i];

  // Reverse exclusive prefix sum: suffix_excl(i) = sum_{j>i} stg[j].
  const int base = 4 * tid;
  float r0 = sstg[NROWS - 1 - (base + 0)];
  float r1 = sstg[NROWS - 1 - (base + 1)];
  float r2 = sstg[NROWS - 1 - (base + 2)];
  float r3 = sstg[NROWS - 1 - (base + 3)];
  float l0 = r0, l1 = l0 + r1, l2 = l1 + r2, l3 = l2 + r3;
  sscan[tid] = l3;
  __syncthreads();
  for (int off = 1; off < 1024; off <<= 1) {
    float v = sscan[tid];
    float w = (tid >= off) ? sscan[tid - off] : 0.0f;
    __syncthreads();
    sscan[tid] = v + w;
    __syncthreads();
  }
  float eb = sscan[tid] - l3;
  float e0 = eb, e1 = eb + l0, e2 = eb + l1, e3 = eb + l2;
  coef[NROWS - 1 - (base + 0)] = fminf(r0, fmaxf(0.0f, 1.0f - e0));
  coef[NROWS - 1 - (base + 1)] = fminf(r1, fmaxf(0.0f, 1.0f - e1));
  coef[NROWS - 1 - (base + 2)] = fminf(r2, fmaxf(0.0f, 1.0f - e2));
  coef[NROWS - 1 - (base + 3)] = fminf(r3, fmaxf(0.0f, 1.0f - e3));
}

#if USE_TDM
// ---------------------------------------------------------------------------
// TDM descriptor (CDNA5 ISA):
//   Group0: w0 = count=1 | is_restore=0 | gather=0
//           w1 = lds byte address
//           w2 = global_addr[31:0], w3 = global_addr[56:32] | type(2)<<30
//   Group1: w0 = workgroup_mask(0) | data_size(2=4B)<<16
//           tensor_dim0 bits79:48, tensor_dim1 bits111:80,
//           tile_dim0 bits127:112, tile_dim1 bits143:128,
//           tensor_dim0_stride bits207:160.
//   2D tile: groups 2/3 zero. OOB rows (y >= tensor_dim1) load as zeros --
//   used to clamp the non-existent prev_Val row 4095, patched from vt.
// ---------------------------------------------------------------------------
__device__ __forceinline__ void tdm_copy(bool is_store, const void* gptr,
                                         void* ldsPtr, unsigned dim1) {
  unsigned long long ga = (unsigned long long)(uintptr_t)gptr;
  v4u g0;
  g0[0] = 1u;
  g0[1] = (unsigned)(uintptr_t)ldsPtr;          // LDS offset (addr[31:0])
  g0[2] = (unsigned)ga;
  g0[3] = (unsigned)(ga >> 32) | (2u << 30);    // type = "image"
  v8i g1;
  g1[0] = 0x00020000;                                            // 4B elems
  g1[1] = (int)(((unsigned)TILE_C & 0xFFFFu) << 16);             // tdim0 lo
  g1[2] = (int)((((unsigned)TILE_C) >> 16) | ((dim1 & 0xFFFFu) << 16));
  g1[3] = (int)((dim1 >> 16) | (((unsigned)TILE_C) << 16));      // tile_dim0
  g1[4] = TILE_R;                                                // tile_dim1
  g1[5] = MCOLS;                                                 // row stride
  g1[6] = 0;
  g1[7] = 0;
  v4i z4 = {0, 0, 0, 0};
#if TDM_6ARG
  v8i z8 = {0, 0, 0, 0, 0, 0, 0, 0};
  if (is_store) __builtin_amdgcn_tensor_store_from_lds(g0, g1, z4, z4, z8, 0);
  else          __builtin_amdgcn_tensor_load_to_lds(g0, g1, z4, z4, z8, 0);
#else
  if (is_store) __builtin_amdgcn_tensor_store_from_lds(g0, g1, z4, z4, 0);
  else          __builtin_amdgcn_tensor_load_to_lds(g0, g1, z4, z4, 0);
#endif
}

// ---------------------------------------------------------------------------
// Kernel B (TDM pipeline): grid (64 col-blocks, 8 K-bands) x 256 threads.
// Wave 0 double-buffers 32x128 tiles: TDM load -> (patch vt row) ->
// TDM store to out_Val + WMMA accumulate from LDS. Per-wave in-order TDM
// execution guarantees store(i) completes before load(i+2) reuses its buffer;
// barriers order LDS reads against reloads.
// WMMA layouts: A 16x4 f32 V0/V1 = coef[k+2*half]/coef[k+2*half+1] (all M rows
// identical); B 4x16 f32 V0/V1 = tile rows k+2*half / k+2*half+1 at col n;
// D row M=0 = rt partial in acc[0], lanes 0-15.
// ---------------------------------------------------------------------------
__global__ void __launch_bounds__(256) fused_copy_wmma(
    const float* __restrict__ prev_Val, const float* __restrict__ vt,
    const float* __restrict__ coef, float* __restrict__ out_Val,
    float* __restrict__ partial) {
  __shared__ __align__(16) float sc[ROWS_PER_SPLIT];
  __shared__ __align__(16) float buf[2][TILE_R * TILE_C];  // 2 x 16KB
  const int tid = threadIdx.x;
  const int ks  = blockIdx.y;
  const int k0  = ks * ROWS_PER_SPLIT;
  const int c0  = blockIdx.x * TILE_C;

  for (int i = tid; i < ROWS_PER_SPLIT; i += 256) sc[i] = coef[k0 + i];

  if (tid < 32) {  // wave 0 issues DMA (TDM ignores EXEC; branch skips it)
    tdm_copy(false, prev_Val + ((size_t)k0 * MCOLS + c0), buf[0],
             (unsigned)(T_PREV - k0));
  }
  __syncthreads();

  const int lane = tid & 31;
  const int wave = tid >> 5;
  const int half = lane >> 4;
  const int col  = c0 + wave * 16 + (lane & 15);
  const int wc   = wave * 16 + (lane & 15);

  v8f acc = {};
  for (int ch = 0; ch < NCHUNK; ++ch) {
    const int row0 = k0 + ch * TILE_R;
    float* cur = buf[ch & 1];

    if (tid < 32) __builtin_amdgcn_s_wait_tensorcnt(0);  // tile landed in LDS
    __syncthreads();

    if (row0 + TILE_R == NROWS) {  // tile contains row 4095: splice in vt
      for (int i = tid; i < TILE_C; i += 256)
        cur[(TILE_R - 1) * TILE_C + i] = vt[c0 + i];
    }
    __syncthreads();

    if (tid < 32) {
      tdm_copy(true, out_Val + ((size_t)row0 * MCOLS + c0), cur,
               (unsigned)(NROWS - row0));
      if (ch + 1 < NCHUNK) {
        const int nr0 = row0 + TILE_R;
        tdm_copy(false, prev_Val + ((size_t)nr0 * MCOLS + c0), buf[(ch + 1) & 1],
                 (unsigned)(T_PREV - nr0));
      }
    }

    for (int kk = 0; kk < TILE_R; kk += 4) {
      v2f a = *(const v2f*)&sc[ch * TILE_R + kk + 2 * half];
      const int rr = kk + 2 * half;
      v2f b;
      b.x = cur[rr * TILE_C + wc];
      b.y = cur[(rr + 1) * TILE_C + wc];
      // 8 args: (neg_a, A, neg_b, B, c_mod, C, reuse_a, reuse_b)
      acc = __builtin_amdgcn_wmma_f32_16x16x4_f32(
          false, a, false, b, (short)0, acc, false, false);
    }
    __syncthreads();  // all waves done reading cur before it is reloaded
  }
  if (half == 0) partial[ks * MCOLS + col] = acc[0];
  // outstanding TDM store drained by S_ENDPGM's implicit wait-idle
}

#else  // ---- fallback: per-lane NT streams (proven path) --------------------
__global__ void __launch_bounds__(256) fused_copy_wmma(
    const float* __restrict__ prev_Val, const float* __restrict__ vt,
    const float* __restrict__ coef, float* __restrict__ out_Val,
    float* __restrict__ partial) {
  __shared__ __align__(16) float sc[ROWS_PER_SPLIT];
  const int tid = threadIdx.x;
  const int ks  = blockIdx.y;
  const int k0  = ks * ROWS_PER_SPLIT;
  for (int i = tid; i < ROWS_PER_SPLIT; i += 256) sc[i] = coef[k0 + i];
  __syncthreads();
  const int lane = tid & 31;
  const int wave = tid >> 5;
  const int half = lane >> 4;
  const int col  = blockIdx.x * 128 + wave * 16 + (lane & 15);
  v8f acc = {};
  for (int kk = 0; kk < ROWS_PER_SPLIT; kk += 4) {
    const int r = k0 + kk + 2 * half;
    v2f a = *(const v2f*)&sc[kk + 2 * half];
    const float* s0 = (r     < T_PREV) ? (prev_Val + (size_t)r * MCOLS)       : vt;
    const float* s1 = (r + 1 < T_PREV) ? (prev_Val + (size_t)(r + 1) * MCOLS) : vt;
    float b0 = __builtin_nontemporal_load(s0 + col);
    float b1 = __builtin_nontemporal_load(s1 + col);
    __builtin_nontemporal_store(b0, out_Val + (size_t)r * MCOLS + col);
    __builtin_nontemporal_store(b1, out_Val + (size_t)(r + 1) * MCOLS + col);
    v2f b; b.x = b0; b.y = b1;
    acc = __builtin_amdgcn_wmma_f32_16x16x4_f32(
        false, a, false, b, (short)0, acc, false, false);
  }
  if (half == 0) partial[ks * MCOLS + col] = acc[0];
}
#endif

// ---------------------------------------------------------------------------
// Kernel C: deterministic fixed-order reduction of the K-split partials.
// ---------------------------------------------------------------------------
__global__ void __launch_bounds__(256) reduce_rt_kernel(
    const float* __restrict__ partial, float* __restrict__ out_rt) {
  const int c = blockIdx.x * 256 + threadIdx.x;
  float s = 0.0f;
#pragma unroll
  for (int j = 0; j < KSPLIT; ++j) s += partial[j * MCOLS + c];
  out_rt[c] = s;
}

extern "C" void kernel_launch(void* const* d_in, const int* in_sizes, int n_in,
                              void* d_out, int out_size, void* d_ws, size_t ws_size,
                              hipStream_t stream) {
  const float* prev_Val = (const float*)d_in[0];  // (4095, 8192)
  const float* prev_stg = (const float*)d_in[1];  // (4095,)
  const float* dt       = (const float*)d_in[2];  // (1,)
  const float* ut       = (const float*)d_in[3];  // (1,)
  const float* vt       = (const float*)d_in[4];  // (1,1,8192)

  float* out     = (float*)d_out;
  float* out_Val = out;                            // 4096*8192
  float* out_stg = out + (size_t)NROWS * MCOLS;    // 4096
  float* out_rt  = out_stg + NROWS;                // 8192

  float* coef    = (float*)d_ws;                   // 4096 floats
  float* partial = coef + NROWS;                   // KSPLIT*8192 floats

  scan_coef_kernel<<<1, 1024, 0, stream>>>(prev_stg, dt, ut, out_stg, coef);

  dim3 gridB(MCOLS / TILE_C, KSPLIT);              // 64 x 8 blocks, 256 thr
  fused_copy_wmma<<<gridB, 256, 0, stream>>>(prev_Val, vt, coef, out_Val, partial);

  reduce_rt_kernel<<<MCOLS / 256, 256, 0, stream>>>(partial, out_rt);
}